// Classifier_Triu_69990787056003
// MI455X (gfx1250) — compile-verified
//
#include <hip/hip_runtime.h>
#include <hip/hip_bf16.h>

// ---------------------------------------------------------------------------
// Problem constants (match reference)
// ---------------------------------------------------------------------------
#define D   64
#define C   64
#define NS  1024
#define NQ  32768
#define REG 0.3f

typedef __attribute__((ext_vector_type(2))) float        v2f;
typedef __attribute__((ext_vector_type(8))) float        v8f;
typedef __attribute__((ext_vector_type(4))) unsigned int v4u;
typedef __attribute__((ext_vector_type(8))) int          v8i_;
typedef __attribute__((ext_vector_type(4))) int          v4i_;

// LDS row stride (dwords) for the staged A_j tile: 64 data + 8 pad
#define APAD 72

// ---------------------------------------------------------------------------
// Workspace layout (floats)
// ---------------------------------------------------------------------------
#define WS_XS     0                     // [NS][D]   normalized support
#define WS_CSUM   (WS_XS + NS*D)        // [C][D]    per-class sums
#define WS_NJ     (WS_CSUM + C*D)       // [C]       per-class counts (float)
#define WS_MU     (WS_NJ + C)           // [C][D]
#define WS_SIGMA  (WS_MU + C*D)         // [C][D][D]
#define WS_A      (WS_SIGMA + C*D*D)    // [C][D][D]
#define WS_AMU    (WS_A + C*D*D)        // [C][D]
#define WS_MUAMU  (WS_AMU + C*D)        // [C]
#define WS_RQ     (WS_MUAMU + C)        // [NQ]      1/||xq|| per row
#define WS_TOTAL  (WS_RQ + NQ)

// ---------------------------------------------------------------------------
// Kernel 1: normalize support rows, accumulate class sums + counts (atomics)
// ---------------------------------------------------------------------------
__global__ void k_support(const float* __restrict__ Xsup,
                          const int*   __restrict__ ysup,
                          float* __restrict__ Xs,
                          float* __restrict__ csum,
                          float* __restrict__ Njf) {
    int n = blockIdx.x * blockDim.x + threadIdx.x;
    if (n >= NS) return;
    float x[D];
    float ss = 0.f;
    #pragma unroll 8
    for (int d = 0; d < D; ++d) { x[d] = Xsup[n * D + d]; ss += x[d] * x[d]; }
    float rn = 1.0f / fmaxf(sqrtf(ss), 1e-12f);
    int c = ysup[n];
    #pragma unroll 8
    for (int d = 0; d < D; ++d) {
        float v = x[d] * rn;
        Xs[n * D + d] = v;
        atomicAdd(&csum[c * D + d], v);
    }
    atomicAdd(&Njf[c], 1.0f);
}

// ---------------------------------------------------------------------------
// Kernel 2: mu[j][d] = head[d] * classMean[j][d]
// ---------------------------------------------------------------------------
__global__ void k_mu(const float* __restrict__ csum,
                     const float* __restrict__ Njf,
                     const float* __restrict__ m,
                     const float* __restrict__ kappa_p,
                     float* __restrict__ mu) {
    int t = blockIdx.x * blockDim.x + threadIdx.x;
    if (t >= C * D) return;
    int j = t / D, d = t % D;
    float kap = kappa_p[0];
    float N0  = Njf[0];
    float head = kap / (kap + N0) * m[d] + N0 / (kap + N0);
    mu[t] = head * csum[t] / Njf[j];
}

// ---------------------------------------------------------------------------
// Kernel 3: per-class covariance sigma_j (one workgroup per class)
// ---------------------------------------------------------------------------
__global__ void k_sigma(const float* __restrict__ Xs,
                        const int*   __restrict__ ysup,
                        const float* __restrict__ mu,
                        const float* __restrict__ S,
                        const float* __restrict__ m,
                        const float* __restrict__ Njf,
                        const float* __restrict__ nu_p,
                        const float* __restrict__ kappa_p,
                        float* __restrict__ sigma) {
    int j   = blockIdx.x;
    int tid = threadIdx.x;          // 256 threads: 4 threads per row d
    int d   = tid >> 2;
    int c0  = (tid & 3) * 16;
    __shared__ float rowsm[D];

    float acc[16];
    #pragma unroll
    for (int i = 0; i < 16; ++i) acc[i] = 0.f;

    for (int n = 0; n < NS; ++n) {
        int yn = ysup[n];           // uniform across block
        if (yn == j) {
            if (tid < D) rowsm[tid] = Xs[n * D + tid];
            __syncthreads();
            float xd = rowsm[d];
            #pragma unroll
            for (int i = 0; i < 16; ++i) acc[i] += xd * rowsm[c0 + i];
            __syncthreads();
        }
    }

    float Njj  = Njf[j];
    float kap  = kappa_p[0];
    float left = 1.0f / (nu_p[0] + Njj + (float)D + 2.0f);
    float md   = m[d];
    float mud  = mu[j * D + d];

    for (int i = 0; i < 16; ++i) {
        int e  = c0 + i;
        int mn = (d < e) ? d : e;
        float mid = 0.f;
        for (int k = 0; k <= mn; ++k) mid += S[d * D + k] * S[e * D + k];
        float val = mid + acc[i] + kap * md * m[e]
                  - (kap + Njj) * mud * mu[j * D + e];
        sigma[j * D * D + d * D + e] = left * val;
    }
}

// ---------------------------------------------------------------------------
// Kernel 4: Gauss-Jordan inversion (partial pivoting) per class
// ---------------------------------------------------------------------------
__global__ void k_invert(const float* __restrict__ sigma,
                         float* __restrict__ A) {
    int j   = blockIdx.x;
    int tid = threadIdx.x;          // 256 threads
    __shared__ float aug[D * 128];  // [64][128] = 32KB
    __shared__ int   piv;
    __shared__ float pinv;

    for (int idx = tid; idx < D * D; idx += 256) {
        int r = idx >> 6, c = idx & 63;
        aug[r * 128 + c]      = sigma[j * D * D + idx];
        aug[r * 128 + 64 + c] = (r == c) ? 1.0f : 0.0f;
    }
    __syncthreads();

    for (int k = 0; k < D; ++k) {
        if (tid == 0) {
            int p = k; float best = fabsf(aug[k * 128 + k]);
            for (int r = k + 1; r < D; ++r) {
                float v = fabsf(aug[r * 128 + k]);
                if (v > best) { best = v; p = r; }
            }
            piv = p;
        }
        __syncthreads();
        if (piv != k && tid < 128) {
            float t = aug[k * 128 + tid];
            aug[k * 128 + tid]   = aug[piv * 128 + tid];
            aug[piv * 128 + tid] = t;
        }
        __syncthreads();
        if (tid == 0) pinv = 1.0f / aug[k * 128 + k];
        __syncthreads();
        if (tid < 128) aug[k * 128 + tid] *= pinv;
        __syncthreads();
        int r  = tid >> 2;
        int cq = (tid & 3) * 32;
        float factor = aug[r * 128 + k];
        __syncthreads();
        if (r != k) {
            #pragma unroll 8
            for (int c = cq; c < cq + 32; ++c)
                aug[r * 128 + c] -= factor * aug[k * 128 + c];
        }
        __syncthreads();
    }

    for (int idx = tid; idx < D * D; idx += 256) {
        int r = idx >> 6, c = idx & 63;
        A[j * D * D + idx] = (1.0f - REG) * aug[r * 128 + 64 + c]
                           + ((r == c) ? REG : 0.0f);
    }
}

// ---------------------------------------------------------------------------
// Kernel 5: Amu[j] = A_j mu_j ; muAmu[j] = mu_j . Amu[j]
// ---------------------------------------------------------------------------
__global__ void k_amu(const float* __restrict__ A,
                      const float* __restrict__ mu,
                      float* __restrict__ Amu,
                      float* __restrict__ muAmu) {
    int j = blockIdx.x, d = threadIdx.x;   // 64 threads
    const float* Aj  = A + j * D * D;
    const float* muj = mu + j * D;
    float s = 0.f;
    #pragma unroll 8
    for (int e = 0; e < D; ++e) s += Aj[d * D + e] * muj[e];
    Amu[j * D + d] = s;
    __shared__ float red[D];
    red[d] = muj[d] * s;
    __syncthreads();
    if (d == 0) {
        float t = 0.f;
        for (int i = 0; i < D; ++i) t += red[i];
        muAmu[j] = t;
    }
}

// ---------------------------------------------------------------------------
// Kernel 6: per-query inverse norms
// ---------------------------------------------------------------------------
__global__ void k_rq(const float* __restrict__ Xq, float* __restrict__ rq) {
    int n = blockIdx.x * blockDim.x + threadIdx.x;
    if (n >= NQ) return;
    float ss = 0.f;
    #pragma unroll 8
    for (int d = 0; d < D; ++d) { float v = Xq[n * D + d]; ss += v * v; }
    rq[n] = 1.0f / fmaxf(sqrtf(ss), 1e-12f);
}

// ---------------------------------------------------------------------------
// TDM: DMA one 64x64 f32 A_j tile (row-major) into LDS with 8-dword row pad
// (LDS row stride becomes 72 dwords -> conflict-free B-fragment reads).
// D# built per CDNA5 ISA ch.8: 2D tensor, data_size=4B, tile=64x64,
// pad_enable, pad_interval=64 dwords (code 5), pad_amount=8 dwords (code 7).
// This toolchain exposes the 6-arg builtin:
//   (uint32x4 g0, int32x8 g1, int32x4 g2, int32x4 g3, int32x8 extra, i32 cpol)
// Groups 2/3 (and the extra group) are zero: tile_dim2==0 marks the tensor 2D.
// ---------------------------------------------------------------------------
__device__ __forceinline__ void tdm_load_A(const float* gsrc, unsigned int lds_off) {
    unsigned long long ga = (unsigned long long)(uintptr_t)gsrc;
    v4u g0;
    g0[0] = 1u;                                             // count=1 (valid)
    g0[1] = lds_off;                                        // lds_addr (bytes)
    g0[2] = (unsigned int)(ga & 0xFFFFFFFFull);             // global_addr[31:0]
    g0[3] = (unsigned int)((ga >> 32) & 0x01FFFFFFull)      // global_addr[56:32]
          | (2u << 30);                                     // type=2 ("image")
    v8i_ g1;
    g1[0] = (int)((2u << 16)       // data_size = 4B
                | (1u << 20)       // pad_enable
                | (5u << 22)       // pad_interval: 64 dwords
                | (7u << 25));     // pad_amount : 8 dwords
    g1[1] = (int)(64u << 16);      // tensor_dim0 = 64
    g1[2] = (int)(64u << 16);      // tensor_dim1 = 64
    g1[3] = (int)(64u << 16);      // tile_dim0   = 64
    g1[4] = 64;                    // tile_dim1   = 64 (tile_dim2 = 0 -> 2D)
    g1[5] = 64;                    // tensor_dim0_stride = 64
    g1[6] = 0;
    g1[7] = 0;
    v4i_ z4 = {0, 0, 0, 0};
    v8i_ z8 = {0, 0, 0, 0, 0, 0, 0, 0};
    __builtin_amdgcn_tensor_load_to_lds(g0, g1, z4, z4, z8, 0);
}

// ---------------------------------------------------------------------------
// Kernel 7: predict — WMMA f32 16x16x4 batched quadratic forms.
// 256 threads = 8 waves; each wave owns a 16-query tile. A_j tiles are
// double-buffered into LDS by the Tensor Data Mover (wave 0 issues, TENSORcnt
// ordered, workgroup barrier publishes).
// out[n][j] = -( x^T A_j x - 2 x.Amu_j + muAmu_j ),  x = normalized query
// ---------------------------------------------------------------------------
__global__ void k_predict(const float* __restrict__ Xq,
                          const float* __restrict__ rq,
                          const float* __restrict__ A,
                          const float* __restrict__ Amu,
                          const float* __restrict__ muAmu,
                          float* __restrict__ out) {
    const int tid  = threadIdx.x;
    const int wid  = tid >> 5;
    const int lane = tid & 31;
    const int lg   = lane >> 4;     // half-group: 0 or 1
    const int ll   = lane & 15;     // lane within group
    const int n0   = blockIdx.x * 128 + wid * 16;   // 16-query tile base

    __shared__ float Asm[2][D * APAD + 8];   // double-buffered A_j (padded rows)
    __shared__ float Amusm[D];

    const unsigned int lds0 = (unsigned int)(uintptr_t)&Asm[0][0];
    const unsigned int lds1 = (unsigned int)(uintptr_t)&Asm[1][0];

    // Kick off DMA of A_0 before doing anything else (overlaps fragment loads).
    if (wid == 0) tdm_load_A(A, lds0);

    // ---- A-operand fragments: 16 chunks of 16x4 (full K=64), held in regs.
    // lane layout: M = ll, K(chunk) = v + 2*lg  ->  global col = 4*kk + v + 2*lg
    v2f a[16];
    {
        int row   = n0 + ll;
        float rn  = rq[row];
        int rbase = row * D;
        #pragma unroll
        for (int kk = 0; kk < 16; ++kk) {
            int cb = 4 * kk + 2 * lg;
            v2f av;
            av[0] = Xq[rbase + cb]     * rn;
            av[1] = Xq[rbase + cb + 1] * rn;
            a[kk] = av;
        }
    }

    // ---- query values in C-layout (row M = v + 8*lg, col N = ll), per e-tile
    float rqv[8];
    float xcs[4][8];
    #pragma unroll
    for (int v = 0; v < 8; ++v) {
        int rr = n0 + v + 8 * lg;
        rqv[v] = rq[rr];
        #pragma unroll
        for (int et = 0; et < 4; ++et)
            xcs[et][v] = Xq[rr * D + et * 16 + ll] * rqv[v];
    }

    for (int j = 0; j < C; ++j) {
        // prefetch next class; wait until the *oldest* TDM (this class) landed
        if (wid == 0) {
            if (j + 1 < C) {
                tdm_load_A(A + (j + 1) * D * D, ((j + 1) & 1) ? lds1 : lds0);
                __builtin_amdgcn_s_wait_tensorcnt((short)1);
            } else {
                __builtin_amdgcn_s_wait_tensorcnt((short)0);
            }
        }
        if (tid < D) Amusm[tid] = Amu[j * D + tid];
        __syncthreads();                      // publish TDM data + Amusm

        const float* __restrict__ AsmJ = &Asm[j & 1][0];
        float mAm = muAmu[j];
        float qsum[8];
        #pragma unroll
        for (int v = 0; v < 8; ++v) qsum[v] = 0.f;

        #pragma unroll
        for (int et = 0; et < 4; ++et) {
            const int ecol = et * 16 + ll;
            float amue = Amusm[ecol];
            v8f acc = {};                     // SRC2 = inline 0

            #pragma unroll
            for (int kk = 0; kk < 16; ++kk) {
                // B operand 4x16: N = ll, K = v + 2*lg -> row = 4*kk + v + 2*lg
                int kb = 4 * kk + 2 * lg;
                v2f b;
                b[0] = AsmJ[(kb + 0) * APAD + ecol];
                b[1] = AsmJ[(kb + 1) * APAD + ecol];
                acc = __builtin_amdgcn_wmma_f32_16x16x4_f32(
                        false, a[kk], false, b, (short)0, acc, false, false);
            }
            // fold the cross term here: x . (A x - 2 Amu)
            #pragma unroll
            for (int v = 0; v < 8; ++v)
                qsum[v] += (acc[v] - 2.0f * amue) * xcs[et][v];
        }
        __syncthreads();   // all LDS reads of buf[j&1] + Amusm done

        // reduce over the 16 columns (lanes within half-group) and store
        #pragma unroll
        for (int v = 0; v < 8; ++v) {
            float r = qsum[v];
            r += __shfl_xor(r, 1, 32);
            r += __shfl_xor(r, 2, 32);
            r += __shfl_xor(r, 4, 32);
            r += __shfl_xor(r, 8, 32);
            if (ll == 0) {
                int rr = n0 + v + 8 * lg;
                out[rr * C + j] = -(r + mAm);
            }
        }
    }
}

// ---------------------------------------------------------------------------
// Launch
// ---------------------------------------------------------------------------
extern "C" void kernel_launch(void* const* d_in, const int* in_sizes, int n_in,
                              void* d_out, int out_size, void* d_ws, size_t ws_size,
                              hipStream_t stream) {
    const float* Xsup   = (const float*)d_in[0];
    const int*   ysup   = (const int*)  d_in[1];
    const float* Xq     = (const float*)d_in[2];
    const float* m      = (const float*)d_in[3];
    const float* S      = (const float*)d_in[4];
    const float* nu_p   = (const float*)d_in[5];
    const float* kappa  = (const float*)d_in[6];
    float* out = (float*)d_out;

    float* W      = (float*)d_ws;
    float* Xs     = W + WS_XS;
    float* csum   = W + WS_CSUM;
    float* Njf    = W + WS_NJ;
    float* mu     = W + WS_MU;
    float* sigma  = W + WS_SIGMA;
    float* A      = W + WS_A;
    float* Amu    = W + WS_AMU;
    float* muAmu  = W + WS_MUAMU;
    float* rq     = W + WS_RQ;

    // zero the atomic accumulators (csum + Njf are contiguous)
    (void)hipMemsetAsync(csum, 0, (C * D + C) * sizeof(float), stream);

    k_support<<<NS / 256, 256, 0, stream>>>(Xsup, ysup, Xs, csum, Njf);
    k_mu     <<<(C * D + 255) / 256, 256, 0, stream>>>(csum, Njf, m, kappa, mu);
    k_sigma  <<<C, 256, 0, stream>>>(Xs, ysup, mu, S, m, Njf, nu_p, kappa, sigma);
    k_invert <<<C, 256, 0, stream>>>(sigma, A);
    k_amu    <<<C, D, 0, stream>>>(A, mu, Amu, muAmu);
    k_rq     <<<NQ / 256, 256, 0, stream>>>(Xq, rq);
    k_predict<<<NQ / 128, 256, 0, stream>>>(Xq, rq, A, Amu, muAmu, out);
}